// FSA_13022340842098
// MI455X (gfx1250) — compile-verified
//
#include <hip/hip_runtime.h>
#include <cstdint>
#include <cstddef>

// Soft-FSA scan for MI455X (gfx1250).
//   S=512 states, I=128 symbols, V=16 outputs, L=4096 steps.
// Roofline: 2.2 GFLOP total; softmaxed T = 128 MB fits the 192 MB L2; each
// step gathers a 1 MB slice T[:,x_t,:] => the scan is L2-bandwidth bound and
// strictly sequential in t. A single WGP can only pull ~128B/clk, so the
// per-step gather is spread across 32 cooperating workgroups (one 16-state
// WMMA tile each, 8 waves splitting K=512), synchronized each step with a
// generation-counter global barrier. GEMV uses exact-precision f32 WMMA
// (V_WMMA_F32_16X16X4_F32); T is pre-swizzled into A-fragment order so each
// lane issues coalesced global_load_b128.

#define S_STATES 512
#define NI       128
#define NV       16
#define NL       4096
#define NBLK     32      // cooperating scan blocks (one per s' tile)

typedef __attribute__((ext_vector_type(2))) float v2f;
typedef __attribute__((ext_vector_type(8))) float v8f;

// ---------------------------------------------------------------------------
// prep_T: softmax each T_param[s, inp, :] row (over s') and scatter into WMMA
// A-fragment order. A-matrix 16x4 f32 layout (ISA 7.12.2):
//   lanes 0-15 : M = lane,    VGPR0 = K0, VGPR1 = K1
//   lanes 16-31: M = lane-16, VGPR0 = K2, VGPR1 = K3
// Chunk pairs (k-chunks 2c, 2c+1) are packed as 4 floats per lane so the scan
// issues one global_load_b128 per lane per 2 WMMAs.
// Tsw layout: [inp][tile(32)][cpair(64)][lane(32)][4 floats]
// Element (s'=m-dim, s=k-dim): A[m=s', k=s] = softmaxT[s][inp][s'].
// ---------------------------------------------------------------------------
__global__ void fsa_prep_T(const float* __restrict__ Tp, float* __restrict__ Tsw) {
  const int row = blockIdx.x;           // row = s*NI + inp
  const int s   = row >> 7;             // NI = 128
  const int inp = row & (NI - 1);
  const int tid = threadIdx.x;          // 256 threads, 2 elements each
  __shared__ float red[256];

  const float* rp = Tp + (size_t)row * S_STATES;
  float a = rp[tid];
  float b = rp[tid + 256];

  red[tid] = fmaxf(a, b);
  __syncthreads();
  for (int off = 128; off > 0; off >>= 1) {
    if (tid < off) red[tid] = fmaxf(red[tid], red[tid + off]);
    __syncthreads();
  }
  const float m = red[0];
  __syncthreads();

  const float ea = __expf(a - m);
  const float eb = __expf(b - m);
  red[tid] = ea + eb;
  __syncthreads();
  for (int off = 128; off > 0; off >>= 1) {
    if (tid < off) red[tid] += red[tid + off];
    __syncthreads();
  }
  const float inv = 1.0f / red[0];

  // k-dim coordinates are fixed for the whole row (k = s)
  const int k       = s;
  const int c       = k >> 2;                 // 4-wide k-chunk
  const int koff    = k & 3;
  const int cpair   = c >> 1;
  const int laneadd = (koff >= 2) ? 16 : 0;   // upper half-wave holds K2,K3
  const int fsel    = ((c & 1) << 1) | (koff & 1);

  #pragma unroll
  for (int e = 0; e < 2; ++e) {
    const int   j    = tid + e * 256;         // j = s' (m-dim)
    const float val  = (e == 0 ? ea : eb) * inv;
    const int   tile = j >> 4;
    const int   mm   = j & 15;
    const size_t idx =
        ((((size_t)inp * 32 + tile) * 64 + cpair) * 32 + (mm + laneadd)) * 4 + fsel;
    Tsw[idx] = val;
  }
}

// ---------------------------------------------------------------------------
// prep_O: softmax each O_param[s, inp, :] row over V=16. One thread per row.
// ---------------------------------------------------------------------------
__global__ void fsa_prep_O(const float* __restrict__ Op, float* __restrict__ Osm) {
  const int r = blockIdx.x * blockDim.x + threadIdx.x;
  if (r >= S_STATES * NI) return;
  const float* p = Op + (size_t)r * NV;
  float v[NV];
  float m = -3.0e38f;
  #pragma unroll
  for (int i = 0; i < NV; ++i) { v[i] = p[i]; m = fmaxf(m, v[i]); }
  float sum = 0.0f;
  #pragma unroll
  for (int i = 0; i < NV; ++i) { v[i] = __expf(v[i] - m); sum += v[i]; }
  const float inv = 1.0f / sum;
  #pragma unroll
  for (int i = 0; i < NV; ++i) Osm[(size_t)r * NV + i] = v[i] * inv;
}

// ---------------------------------------------------------------------------
// prep_init: softmax(init_param) -> states[0]; also reset the step barrier.
// ---------------------------------------------------------------------------
__global__ void fsa_prep_init(const float* __restrict__ ip, float* __restrict__ st0,
                              int* __restrict__ syncw) {
  __shared__ float red[S_STATES];
  const int t = threadIdx.x;
  if (t < 2) syncw[t] = 0;   // cnt, gen
  const float x = ip[t];
  red[t] = x;
  __syncthreads();
  for (int off = 256; off > 0; off >>= 1) {
    if (t < off) red[t] = fmaxf(red[t], red[t + off]);
    __syncthreads();
  }
  const float m = red[0];
  __syncthreads();
  const float e = __expf(x - m);
  red[t] = e;
  __syncthreads();
  for (int off = 256; off > 0; off >>= 1) {
    if (t < off) red[t] += red[t + off];
    __syncthreads();
  }
  st0[t] = e / red[0];
}

// ---------------------------------------------------------------------------
// scan: 32 cooperating blocks (256 threads = 8 wave32 each). Block w owns
// output tile s' in [16w, 16w+16); its 8 waves split the K=512 contraction
// (8 chunk-pairs each => 16 f32 WMMAs/wave/step, two accumulator chains).
//   D[m,n] = sum_k A[m,k]*B[k,n];  A = T fragment, B[k,n] = state[k] (all n).
// B 4x16 f32 layout: lanes 0-15 hold K0(VGPR0),K1(VGPR1); lanes 16-31 K2,K3.
// All 16 D columns are identical; lanes 0 and 16 own M=0..7 / M=8..15.
// K-partials are reduced through LDS; wave 0 stores the 16-float tile with
// agent-scope stores, then arrives at a generation-counter device barrier.
// ---------------------------------------------------------------------------
__global__ void __launch_bounds__(256) fsa_scan(const int* __restrict__ seq,
                                                const float* __restrict__ Tsw,
                                                float* __restrict__ states,
                                                int* __restrict__ syncw) {
  __shared__ float st[S_STATES];
  __shared__ float part[8][16];
  const int tid    = threadIdx.x;
  const int lane   = tid & 31;
  const int kw     = tid >> 5;             // wave id 0..7 = K splitter
  const int w      = blockIdx.x;           // s' tile 0..31
  const int lhalf2 = (lane >> 4) << 1;     // 0 for K0/K1 lanes, 2 for K2/K3 lanes
  const int kb     = kw * 8;               // first chunk-pair for this wave
  int* cnt = syncw;
  int* gen = syncw + 1;

  for (int t = 0; t < NL; ++t) {
    // Reload state_t (written by all 32 blocks). Agent-scope loads bypass the
    // read-only WGP$ so cross-WGP tile stores from the previous step are seen.
    st[tid] = __hip_atomic_load(states + (size_t)t * S_STATES + tid,
                                __ATOMIC_RELAXED, __HIP_MEMORY_SCOPE_AGENT);
    st[tid + 256] = __hip_atomic_load(states + (size_t)t * S_STATES + tid + 256,
                                      __ATOMIC_RELAXED, __HIP_MEMORY_SCOPE_AGENT);
    __syncthreads();

    const int inp = seq[t];
    const float4* lb =
        (const float4*)(Tsw) + ((((size_t)inp * 32 + w) * 64 + kb) * 32 + lane);

    // L2 prefetch of next step's slice (global_prefetch_b8)
    if (t + 1 < NL) {
      const int ninp = seq[t + 1];
      __builtin_prefetch(
          (const float4*)(Tsw) + ((((size_t)ninp * 32 + w) * 64 + kb) * 32 + lane),
          0, 3);
    }

    v8f acc0 = {};
    v8f acc1 = {};
    #pragma unroll
    for (int i = 0; i < 8; ++i) {
      const int    cp = kb + i;
      const float4 tt = lb[(size_t)i * 32];                        // 2 k-chunks of T
      const float2 b0 = *(const float2*)(&st[cp * 8 + lhalf2]);     // state[8cp+{0,1|2,3}]
      const float2 b1 = *(const float2*)(&st[cp * 8 + 4 + lhalf2]); // state[8cp+4+{..}]
      v2f A0 = {tt.x, tt.y};
      v2f A1 = {tt.z, tt.w};
      v2f B0 = {b0.x, b0.y};
      v2f B1 = {b1.x, b1.y};
      acc0 = __builtin_amdgcn_wmma_f32_16x16x4_f32(false, A0, false, B0,
                                                   (short)0, acc0, false, false);
      acc1 = __builtin_amdgcn_wmma_f32_16x16x4_f32(false, A1, false, B1,
                                                   (short)0, acc1, false, false);
    }
    v8f acc = acc0 + acc1;

    // K-partials -> LDS (lane 0: M=0..7, lane 16: M=8..15)
    if ((lane & 15) == 0) {
      const int mb = (lane >> 4) << 3;
      float4 lo = {acc[0], acc[1], acc[2], acc[3]};
      float4 hi = {acc[4], acc[5], acc[6], acc[7]};
      *(float4*)(&part[kw][mb])     = lo;
      *(float4*)(&part[kw][mb + 4]) = hi;
    }
    __syncthreads();

    // Wave 0 reduces the 8 K-partials and publishes the tile (agent scope).
    if (tid < 16) {
      float sv = 0.0f;
      #pragma unroll
      for (int k2 = 0; k2 < 8; ++k2) sv += part[k2][tid];
      __hip_atomic_store(states + (size_t)(t + 1) * S_STATES + w * 16 + tid, sv,
                         __ATOMIC_RELAXED, __HIP_MEMORY_SCOPE_AGENT);
    }

    // Device-wide step barrier. tid 0..15 are in wave 0, so tid 0's fence
    // (s_wait_storecnt) covers the tile stores above before the arrival.
    if (tid == 0) {
      __threadfence();
      const int prev = __hip_atomic_fetch_add(cnt, 1, __ATOMIC_ACQ_REL,
                                              __HIP_MEMORY_SCOPE_AGENT);
      if (prev == NBLK - 1) {
        __hip_atomic_store(cnt, 0, __ATOMIC_RELAXED, __HIP_MEMORY_SCOPE_AGENT);
        __hip_atomic_fetch_add(gen, 1, __ATOMIC_RELEASE, __HIP_MEMORY_SCOPE_AGENT);
      } else {
        while (__hip_atomic_load(gen, __ATOMIC_ACQUIRE,
                                 __HIP_MEMORY_SCOPE_AGENT) <= t) {
          __builtin_amdgcn_s_sleep(2);
        }
      }
    }
    __syncthreads();
  }
}

// ---------------------------------------------------------------------------
// out: out[t][v] = sum_s states[t][s] * Osm[s][x_t][v]; parallel over t.
// 256 threads: v = tid&15, group g = tid>>4 covers 32 states; tree-reduce.
// ---------------------------------------------------------------------------
__global__ void fsa_out(const int* __restrict__ seq,
                        const float* __restrict__ states,
                        const float* __restrict__ Osm,
                        float* __restrict__ out) {
  const int t   = blockIdx.x;
  const int tid = threadIdx.x;
  const int v   = tid & 15;
  const int g   = tid >> 4;
  const int inp = seq[t];
  __shared__ float red[256];

  const float* stp = states + (size_t)t * S_STATES;
  float p = 0.0f;
  #pragma unroll 4
  for (int s = g * 32; s < g * 32 + 32; ++s)
    p += stp[s] * Osm[((size_t)s * NI + inp) * NV + v];
  red[tid] = p;
  __syncthreads();
  for (int off = 128; off >= 16; off >>= 1) {
    if (tid < off) red[tid] += red[tid + off];
    __syncthreads();
  }
  if (tid < 16) out[(size_t)t * NV + tid] = red[tid];
}

// ---------------------------------------------------------------------------
extern "C" void kernel_launch(void* const* d_in, const int* in_sizes, int n_in,
                              void* d_out, int out_size, void* d_ws, size_t ws_size,
                              hipStream_t stream) {
  (void)in_sizes; (void)n_in; (void)out_size; (void)ws_size;
  const int*   seq = (const int*)d_in[0];    // input_seq (L)
  const float* Tp  = (const float*)d_in[1];  // T_param (S,I,S)
  const float* Op  = (const float*)d_in[2];  // O_param (S,I,V)
  const float* ip  = (const float*)d_in[3];  // init_param (S)

  char* ws = (char*)d_ws;
  const size_t tsw_bytes = (size_t)S_STATES * NI * S_STATES * sizeof(float); // 128 MB
  const size_t osm_bytes = (size_t)S_STATES * NI * NV * sizeof(float);       // 4 MB
  const size_t st_bytes  = (size_t)(NL + 1) * S_STATES * sizeof(float);      // 8.4 MB
  float* Tsw    = (float*)ws;
  float* Osm    = (float*)(ws + tsw_bytes);
  float* states = (float*)(ws + tsw_bytes + osm_bytes);
  int*   syncw  = (int*)(ws + tsw_bytes + osm_bytes + st_bytes);             // cnt, gen

  fsa_prep_T<<<S_STATES * NI, 256, 0, stream>>>(Tp, Tsw);
  fsa_prep_O<<<(S_STATES * NI + 255) / 256, 256, 0, stream>>>(Op, Osm);
  fsa_prep_init<<<1, S_STATES, 0, stream>>>(ip, states, syncw);
  fsa_scan<<<NBLK, 256, 0, stream>>>(seq, Tsw, states, syncw);
  fsa_out<<<NL, 256, 0, stream>>>(seq, states, Osm, (float*)d_out);
}